// InterpolationModel_54760833024018
// MI455X (gfx1250) — compile-verified
//
#include <hip/hip_runtime.h>

#ifndef __has_builtin
#define __has_builtin(x) 0
#endif

#if __has_builtin(__builtin_amdgcn_global_load_async_to_lds_b128)
#define USE_ASYNC_LDS 1
#endif

// 128-bit vector type matching the builtin's parameter type, plus
// address-space-qualified pointer typedefs (attribute applies to pointee).
typedef int v4i __attribute__((ext_vector_type(4)));
typedef __attribute__((address_space(1))) v4i* gptr_v4i;  // global
typedef __attribute__((address_space(3))) v4i* lptr_v4i;  // LDS

// ---- async global -> LDS staging helpers -----------------------------------

__device__ __forceinline__ void copy_b128_g2l(void* lds, const void* g) {
#ifdef USE_ASYNC_LDS
  __builtin_amdgcn_global_load_async_to_lds_b128(
      (gptr_v4i)(g), (lptr_v4i)(lds), /*imm offset=*/0, /*imm cpol=*/0);
#else
  *reinterpret_cast<float4*>(lds) = *reinterpret_cast<const float4*>(g);
#endif
}

template <int N>
__device__ __forceinline__ void wait_async() {
#ifdef USE_ASYNC_LDS
#if __has_builtin(__builtin_amdgcn_s_wait_asynccnt)
  __builtin_amdgcn_s_wait_asynccnt(N);
#else
  asm volatile("s_wait_asynccnt %0" ::"i"(N) : "memory");
#endif
#endif
}

// ---- kernel ----------------------------------------------------------------
// x, mask, out: [B=64, T=2048, K=64, C=4]  (KC = 256 contiguous series lanes)
// grid = (NCHUNK=16, B=64), block = 256 threads (8 wave32).
// Block (chunk, b): stages tiles of 16 timesteps (16KB x + 16KB mask,
// contiguous) into LDS via async b128 loads, double buffered. Each thread owns
// one (k,c) series and runs the prev-valid scan over its chunk; chunk-boundary
// state is recovered by short backward/forward walks in global memory.

__global__ __launch_bounds__(256) void interp_scan_kernel(
    const float* __restrict__ x, const int* __restrict__ mask,
    float* __restrict__ out) {
  constexpr int Tn = 2048;
  constexpr int KC = 256;          // K*C contiguous lanes
  constexpr int NCHUNK = 16;
  constexpr int CH = Tn / NCHUNK;  // 128 timesteps per block
  constexpr int TT = 16;           // timesteps per LDS tile
  constexpr int NTILES = CH / TT;  // 8
  constexpr int TILE_E = TT * KC;  // 4096 elements = 16KB

  __shared__ __align__(16) float xs[2][TILE_E];
  __shared__ __align__(16) int   ms[2][TILE_E];

  const int tid   = threadIdx.x;
  const int chunk = blockIdx.x;
  const int b     = blockIdx.y;
  const int t0    = chunk * CH;
  const int t1    = t0 + CH;

  const size_t baseB = (size_t)b * Tn * KC;
  const float* xg = x    + baseB;
  const int*   mg = mask + baseB;
  float*       og = out  + baseB;

  const float* xser = xg + tid;  // this thread's series (stride KC along t)
  const int*   mser = mg + tid;
  float*       oser = og + tid;

  auto issue = [&](int ti, int buf) {
    const size_t off = (size_t)(t0 + ti * TT) * KC;
    const float* xsrc = xg + off;
    const int*   msrc = mg + off;
#pragma unroll
    for (int r = 0; r < 4; ++r) {
      const int e = (r * 256 + tid) * 4;  // 4 dwords per b128
      copy_b128_g2l(&xs[buf][e], xsrc + e);
      copy_b128_g2l(&ms[buf][e], msrc + e);
    }
  };

  // ---- recover previous-valid state from earlier chunks (expected ~2 reads)
  int prev_t = -1;
  float pv = 0.f;
  for (int u = t0 - 1; u >= 0; --u) {
    if (mser[(size_t)u * KC] != 0) {
      prev_t = u;
      pv = xser[(size_t)u * KC];
      break;
    }
  }

  issue(0, 0);

  for (int ti = 0; ti < NTILES; ++ti) {
    const int buf = ti & 1;
    if (ti + 1 < NTILES) {
      issue(ti + 1, buf ^ 1);
      wait_async<8>();   // previous tile's 8 per-wave async ops retired
    } else {
      wait_async<0>();
    }
    __syncthreads();

    const int tbase = t0 + ti * TT;
#pragma unroll 4
    for (int tt = 0; tt < TT; ++tt) {
      const int e = tt * KC + tid;
      const float v = xs[buf][e];
      const int   m = ms[buf][e];
      if (m != 0) {
        const int t = tbase + tt;
        int lo = prev_t + 1;
        if (lo < t0) lo = t0;  // positions < t0 are owned by earlier chunk
        if (prev_t >= 0) {
          const float d = (float)(t - prev_t);
          for (int u = lo; u < t; ++u)
            oser[(size_t)u * KC] =
                pv + (v - pv) * ((float)(u - prev_t) / d);
        } else {
          for (int u = lo; u < t; ++u)  // flat-left extrapolation
            oser[(size_t)u * KC] = v;
        }
        oser[(size_t)t * KC] = v;
        prev_t = t;
        pv = v;
      }
    }
    __syncthreads();
  }

  // ---- trailing gap inside this chunk: need next-valid from later chunks
  int lo = prev_t + 1;
  if (lo < t0) lo = t0;
  if (lo < t1) {
    int nt = -1;
    float nv = 0.f;
    for (int u = t1; u < Tn; ++u) {
      if (mser[(size_t)u * KC] != 0) {
        nt = u;
        nv = xser[(size_t)u * KC];
        break;
      }
    }
    if (prev_t >= 0 && nt >= 0) {
      const float d = (float)(nt - prev_t);
      for (int u = lo; u < t1; ++u)
        oser[(size_t)u * KC] = pv + (nv - pv) * ((float)(u - prev_t) / d);
    } else {
      const float f = (prev_t >= 0) ? pv : ((nt >= 0) ? nv : 0.f);
      for (int u = lo; u < t1; ++u) oser[(size_t)u * KC] = f;
    }
  }
}

// ---- launch ----------------------------------------------------------------
extern "C" void kernel_launch(void* const* d_in, const int* in_sizes, int n_in,
                              void* d_out, int out_size, void* d_ws,
                              size_t ws_size, hipStream_t stream) {
  (void)in_sizes; (void)n_in; (void)d_ws; (void)ws_size; (void)out_size;
  const float* x    = (const float*)d_in[0];
  const int*   mask = (const int*)d_in[1];
  float*       out  = (float*)d_out;

  dim3 grid(16, 64);  // (T-chunks, B)
  interp_scan_kernel<<<grid, 256, 0, stream>>>(x, mask, out);
}